// Long_Information_36567351558726
// MI455X (gfx1250) — compile-verified
//
#include <hip/hip_runtime.h>
#include <hip/hip_bf16.h>

typedef float v2f __attribute__((ext_vector_type(2)));
typedef float v8f __attribute__((ext_vector_type(8)));

// Packed fp32 FMA (VOP3P V_PK_FMA_F32): d = a*b + c on a float2 per lane.
// Guaranteed selection via inline asm (the backend scalarized the
// <2 x float> __builtin_elementwise_fma form). Non-volatile: pure value op,
// scheduler may move/dedup it freely.
static __device__ __forceinline__ v2f pk_fma(v2f a, v2f b, v2f c) {
  v2f d;
  asm("v_pk_fma_f32 %0, %1, %2, %3" : "=v"(d) : "v"(a), "v"(b), "v"(c));
  return d;
}

// ---------------------------------------------------------------------------
// WMMA GEMM + bias:  Y[M,NCOL] = X[M,K] @ W[K,NCOL] + bias[NCOL]
// fp32 end-to-end via V_WMMA_F32_16X16X4_F32. One wave per 16x16 output tile.
// K, NCOL compile-time so all row strides fold into immediate offsets.
// A 16x4 layout: lane L -> row M=L&15, a.x = K=kb, a.y = K=kb+1, kb=2*(L>=16)
// B 4x16 layout: lane L -> col N=L&15, b.x = row K=kb, b.y = row K=kb+1
// C/D: VGPR r, lane L -> M = r + 8*(L>=16), N = L&15
// ---------------------------------------------------------------------------
template <int K, int NCOL>
__global__ void wmma_rootgemm(const float* __restrict__ X,
                              const float* __restrict__ W,
                              const float* __restrict__ bias,
                              float* __restrict__ Y,
                              int M) {
  const int wave = (int)((blockIdx.x * blockDim.x + threadIdx.x) >> 5);
  const int lane = (int)(threadIdx.x & 31);
  constexpr int tilesN = NCOL >> 4;
  const int tm = (wave / tilesN) << 4;   // uniform per wave
  const int tn = (wave % tilesN) << 4;
  if (tm >= M) return;                   // whole-wave uniform exit (EXEC stays all-1s)

  const int mn = lane & 15;              // A-row and B/D-column index
  const int kb = (lane >> 4) << 1;       // 0 for lanes 0-15, 2 for lanes 16-31

  v8f c;
  const float bval = bias[tn + mn];      // per-output-column bias seeds C
#pragma unroll
  for (int r = 0; r < 8; ++r) c[r] = bval;

  const float* __restrict__ Xrow = X + (size_t)(tm + mn) * K + kb;
  const float* __restrict__ Wcol = W + (size_t)kb * NCOL + tn + mn;
#pragma unroll
  for (int k = 0; k < K; k += 4) {
    v2f a, b;
    a.x = Xrow[k];
    a.y = Xrow[k + 1];
    b.x = Wcol[(size_t)k * NCOL];
    b.y = Wcol[(size_t)(k + 1) * NCOL];
    // 8 args: (neg_a, A, neg_b, B, c_mod, C, reuse_a, reuse_b)
    c = __builtin_amdgcn_wmma_f32_16x16x4_f32(false, a, false, b,
                                              (short)0, c, false, false);
  }

  const int mhalf = (lane >> 4) << 3;    // +8 rows for upper half-wave
  float* __restrict__ Yrow = Y + (size_t)(tm + mhalf) * NCOL + tn + mn;
#pragma unroll
  for (int r = 0; r < 8; ++r)
    Yrow[(size_t)r * NCOL] = c[r];       // constant offsets -> store clause
}

// ---------------------------------------------------------------------------
// Fused NNConv message pass (CIN inputs -> 64 outputs):
//   per edge e: W[e] = relu(ea0*A[0,:] + ea1*A[1,:] + b)  (CIN x 64, in-register)
//               msg  = h[src[e]] @ W[e]                   (1x64)
//               out[dst[e], :] += msg                     (atomicAdd = segment_sum)
// Weights staged once per block in LDS (each edge reads all 3*CIN*64 floats, so
// this removes ~E*48KB of L2 traffic). One wave per edge; each lane owns an
// adjacent pair of output channels so all FMAs are packed v_pk_fma_f32
// (2x fp32 throughput per VALU issue).
// ---------------------------------------------------------------------------
template <int CIN>
__global__ void edge_msg(const long long* __restrict__ ei,   // [2, E] int64
                         const float* __restrict__ ea,       // [E, 2]
                         const float* __restrict__ A,        // [2, CIN*64]
                         const float* __restrict__ bvec,     // [CIN*64]
                         const float* __restrict__ hin,      // [N, CIN]
                         float* __restrict__ out,            // [N, 64] (accumulated)
                         int E_) {
  constexpr int Wsz = CIN * 64;
  extern __shared__ float lds[];
  float* a0 = lds;            // Wsz floats
  float* a1 = lds + Wsz;      // Wsz floats
  float* bb = lds + 2 * Wsz;  // Wsz floats
  for (int i = threadIdx.x; i < Wsz; i += blockDim.x) {
    a0[i] = A[i];
    a1[i] = A[Wsz + i];
    bb[i] = bvec[i];
  }
  __syncthreads();

  const int lane = (int)(threadIdx.x & 31);
  const int wavesPerBlock = (int)(blockDim.x >> 5);
  const int gwave = (int)blockIdx.x * wavesPerBlock + (int)(threadIdx.x >> 5);
  const int nwaves = (int)gridDim.x * wavesPerBlock;

  // This lane's pair of output channels (o0, o0+1); o0 even -> 8B-aligned LDS.
  const int o0 = lane << 1;
  const v2f* a0v = (const v2f*)a0;   // indexed in float2 units: i*32 + lane
  const v2f* a1v = (const v2f*)a1;
  const v2f* bbv = (const v2f*)bb;

  for (int e = gwave; e < E_; e += nwaves) {
    const int src = (int)ei[e];
    const int dst = (int)ei[E_ + e];
    const float e0 = ea[2 * e + 0];
    const float e1 = ea[2 * e + 1];
    const v2f e0v = {e0, e0};
    const v2f e1v = {e1, e1};
    const v2f* xs = (const v2f*)(hin + (size_t)src * CIN);  // CIN even, 8B aligned

    v2f acc = {0.f, 0.f};
#pragma unroll 8
    for (int j = 0; j < CIN / 2; ++j) {
      const v2f xv = xs[j];            // two source features, wave-broadcast load
      // i = 2j
      {
        const int idx = (2 * j) * 32 + lane;           // float2 units
        v2f w = pk_fma(e1v, a1v[idx], bbv[idx]);
        w = pk_fma(e0v, a0v[idx], w);
        w.x = fmaxf(w.x, 0.f);                         // ReLU (no pk max f32)
        w.y = fmaxf(w.y, 0.f);
        const v2f xb = {xv.x, xv.x};
        acc = pk_fma(xb, w, acc);
      }
      // i = 2j+1
      {
        const int idx = (2 * j + 1) * 32 + lane;
        v2f w = pk_fma(e1v, a1v[idx], bbv[idx]);
        w = pk_fma(e0v, a0v[idx], w);
        w.x = fmaxf(w.x, 0.f);
        w.y = fmaxf(w.y, 0.f);
        const v2f xb = {xv.y, xv.y};
        acc = pk_fma(xb, w, acc);
      }
    }
    atomicAdd(out + (size_t)dst * 64 + o0,     acc.x);
    atomicAdd(out + (size_t)dst * 64 + o0 + 1, acc.y);
  }
}

extern "C" void kernel_launch(void* const* d_in, const int* in_sizes, int n_in,
                              void* d_out, int out_size, void* d_ws, size_t ws_size,
                              hipStream_t stream) {
  const float*     x     = (const float*)d_in[0];
  const long long* ei    = (const long long*)d_in[1];  // int64 edge_index [2,E]
  const float*     ea    = (const float*)d_in[2];
  const float*     A1    = (const float*)d_in[3];
  const float*     b1    = (const float*)d_in[4];
  const float*     A2    = (const float*)d_in[5];
  const float*     b2    = (const float*)d_in[6];
  const float*     root1 = (const float*)d_in[7];
  const float*     bias1 = (const float*)d_in[8];
  const float*     root2 = (const float*)d_in[9];
  const float*     bias2 = (const float*)d_in[10];
  float* out = (float*)d_out;
  float* h1  = (float*)d_ws;              // [N,64] fp32 intermediate = 4 MB

  const int N_ = 16384;
  const int E_ = in_sizes[1] / 2;         // 65536
  const int C  = 64;

  // WMMA GEMM grid: (N/16)*(C/16) = 4096 tiles, 8 waves (tiles) per 256-thr block
  const int tiles = (N_ / 16) * (C / 16);
  const int gemmBlocks = (tiles * 32) / 256;

  // 1) h1 = x @ root1 + bias1            (writes every element of h1)
  wmma_rootgemm<8, 64><<<gemmBlocks, 256, 0, stream>>>(x, root1, bias1, h1, N_);

  // 2) h1 += segment_sum(relu-edge-MLP messages, layer 1)   LDS = 6 KB
  edge_msg<8><<<512, 256, 3 * 8 * 64 * sizeof(float), stream>>>(
      ei, ea, A1, b1, x, h1, E_);

  // 3) out = h1 @ root2 + bias2          (writes every element of out)
  wmma_rootgemm<64, 64><<<gemmBlocks, 256, 0, stream>>>(h1, root2, bias2, out, N_);

  // 4) out += segment_sum(messages, layer 2)                LDS = 48 KB
  edge_msg<64><<<512, 256, 3 * 64 * 64 * sizeof(float), stream>>>(
      ei, ea, A2, b2, h1, out, E_);
}